// CropLayer_87522843560309
// MI455X (gfx1250) — compile-verified
//
#include <hip/hip_runtime.h>

typedef float v4f __attribute__((ext_vector_type(4)));

namespace {
constexpr int kCrop         = 28;
constexpr int kBoxesPerImg  = 100;
constexpr int kH            = 512;
constexpr int kW            = 512;
constexpr int kC            = 32;
constexpr int kNBox         = 800;
constexpr int kRowsPerBlock = 7;                 // 28 / 4
constexpr int kRowStride4   = kCrop * (kC / 4);  // 224 float4s per output row
}

// Grid: (28/7 = 4 row-groups, 800 boxes). Block: 224 threads = 28 x-positions
// * 8 float4 channel-groups = 7 wave32s. Each block produces 7 output rows of
// one box, amortizing the per-wave setup (box load, IEEE divides, the whole
// per-lane x pipeline) over 7x the output, and exposing 28 independent b128
// gathers per lane for deep memory-level parallelism.
__global__ __launch_bounds__(224) void crop_resize_bilinear(
    const float* __restrict__ images,   // [8,512,512,32]
    const float* __restrict__ boxes,    // [800,4] (y1,x1,y2,x2)
    float*       __restrict__ out)      // [800,28,28,32]
{
    const int tid    = (int)threadIdx.x;              // 0..223
    const int c4     = tid & 7;                       // float4 index in channels
    const int x      = tid >> 3;                      // 0..27
    const int ybase  = (int)blockIdx.x * kRowsPerBlock;
    const int n      = (int)blockIdx.y;               // box 0..799 (uniform)
    const int b      = n / kBoxesPerImg;              // image 0..7  (uniform)

    // Uniform box read -> s_load_b128.
    const float by1 = boxes[n * 4 + 0];
    const float bx1 = boxes[n * 4 + 1];
    const float by2 = boxes[n * 4 + 2];
    const float bx2 = boxes[n * 4 + 3];

    // Exact reference expression order (true f32 divides, once per wave).
    const float hs = (by2 - by1) * (float)(kH - 1) / (float)(kCrop - 1);
    const float ws = (bx2 - bx1) * (float)(kW - 1) / (float)(kCrop - 1);

    // ---- Loop-invariant per-lane x pipeline (computed once) ----
    const float in_x = bx1 * (float)(kW - 1) + (float)x * ws;
    const bool  vx   = (in_x >= 0.0f) && (in_x <= (float)(kW - 1));
    const float x0f  = floorf(in_x);
    const float lx   = in_x - x0f;
    int xi0 = min(max((int)x0f, 0), kW - 1);
    int xi1 = min(xi0 + 1, kW - 1);
    const int off0 = xi0 * (kC / 4) + c4;   // float4-unit gather offsets
    const int off1 = xi1 * (kC / 4) + c4;

    const v4f* __restrict__ img4 =
        (const v4f*)images + (size_t)b * kH * kW * (kC / 4);
    v4f* __restrict__ outp =
        (v4f*)out + ((size_t)n * kCrop + (size_t)ybase) * kRowStride4 + tid;

    const v4f zero = {0.0f, 0.0f, 0.0f, 0.0f};

#pragma unroll
    for (int i = 0; i < kRowsPerBlock; ++i) {
        const int   y    = ybase + i;
        const float in_y = by1 * (float)(kH - 1) + (float)y * hs;
        const bool  vy   = (in_y >= 0.0f) && (in_y <= (float)(kH - 1));
        const float y0f  = floorf(in_y);
        const float ly   = in_y - y0f;
        int yi0 = min(max((int)y0f, 0), kH - 1);
        int yi1 = min(yi0 + 1, kH - 1);

        const v4f* __restrict__ rowT = img4 + (size_t)yi0 * (kW * (kC / 4));
        const v4f* __restrict__ rowB = img4 + (size_t)yi1 * (kW * (kC / 4));

        // 4 independent b128 gathers per row; unroll lets the scheduler
        // hoist all 28 of the block's gathers ahead of the waits.
        const v4f tl = rowT[off0];
        const v4f tr = rowT[off1];
        const v4f bl = rowB[off0];
        const v4f br = rowB[off1];

        const v4f top   = tl + (tr - tl) * lx;
        const v4f bot   = bl + (br - bl) * lx;
        const v4f blend = top + (bot - top) * ly;
        const v4f res   = (vy && vx) ? blend : zero;   // branchless cndmask

        // Per-iteration displacement folds into the store's constant ioffset.
        __builtin_nontemporal_store(res, outp + i * kRowStride4);
    }
}

extern "C" void kernel_launch(void* const* d_in, const int* in_sizes, int n_in,
                              void* d_out, int out_size, void* d_ws, size_t ws_size,
                              hipStream_t stream) {
    const float* images = (const float*)d_in[0];
    const float* boxes  = (const float*)d_in[1];
    float* out          = (float*)d_out;

    dim3 grid(kCrop / kRowsPerBlock, kNBox);   // (4, 800)
    dim3 block(kCrop * (kC / 4));              // 224 threads = 7 wave32s
    crop_resize_bilinear<<<grid, block, 0, stream>>>(images, boxes, out);
}